// GatedLightningIndexer_47038481826267
// MI455X (gfx1250) — compile-verified
//
#include <hip/hip_runtime.h>
#include <math.h>

typedef __attribute__((ext_vector_type(2))) float v2f;
typedef __attribute__((ext_vector_type(8))) float v8f;

#define D_MODEL 2048
#define D_IDX   64
#define N_HEADS 4
#define SEQ     4096
#define BATCH   2
#define ROWS    (BATCH * SEQ)          // 8192 flat rows of hidden_states
#define N_PROJ  (N_HEADS * D_IDX + D_IDX + N_HEADS)  // 256 + 64 + 4 = 324
#define COL_TILES 21                   // ceil(324/16)
#define QK_SCALE 0.125f                // 1/sqrt(64)

__device__ __forceinline__ float sigmoidf_fast(float x) {
    return 1.0f / (1.0f + __expf(-x));
}

// ---------------------------------------------------------------------------
// Kernel 1: fused projection GEMM  out = X @ [Wq;Wk;Ww]^T  via f32 WMMA.
// One wave computes one 16x16 tile. Column n: [0,256)->Q, [256,320)->K,
// [320,324)->gate weights (bias + sigmoid fused).
// ---------------------------------------------------------------------------
__global__ void __launch_bounds__(256)
gli_proj_kernel(const float* __restrict__ X,
                const float* __restrict__ Wq,
                const float* __restrict__ Wk,
                const float* __restrict__ Ww,
                const float* __restrict__ bw,
                float* __restrict__ Qws,   // [ROWS, 256]
                float* __restrict__ Kws,   // [ROWS, 64]
                float* __restrict__ Gws)   // [ROWS, 4]
{
    const int wave    = blockIdx.x * (blockDim.x >> 5) + (threadIdx.x >> 5);
    const int lane    = threadIdx.x & 31;
    const int laneM   = lane & 15;
    const int laneHi  = lane >> 4;

    const int rowTile = wave / COL_TILES;
    const int colTile = wave % COL_TILES;
    if (rowTile >= ROWS / 16) return;

    const int arow = rowTile * 16 + laneM;       // A: M = lane%16
    const int n    = colTile * 16 + laneM;       // B/C column = lane%16

    // Per-lane pointer to the weight row for column n (B = W^T tile).
    const float* wrow;
    if      (n < 256)    wrow = Wq + (size_t)n * D_MODEL;
    else if (n < 320)    wrow = Wk + (size_t)(n - 256) * D_MODEL;
    else if (n < N_PROJ) wrow = Ww + (size_t)(n - 320) * D_MODEL;
    else                 wrow = Ww;              // dummy (result discarded)

    const float* xrow = X + (size_t)arow * D_MODEL;

    v8f acc = {};
    for (int k0 = 0; k0 < D_MODEL; k0 += 4) {
        const int kb = k0 + 2 * laneHi;          // lanes 0-15: K pair {0,1}; 16-31: {2,3}
        v2f a  = *(const v2f*)(xrow + kb);
        v2f bm = *(const v2f*)(wrow + kb);
        acc = __builtin_amdgcn_wmma_f32_16x16x4_f32(
            /*neg_a=*/false, a, /*neg_b=*/false, bm,
            /*c_mod=*/(short)0, acc, /*reuse_a=*/false, /*reuse_b=*/false);
    }

    // C/D layout: VGPR i -> row (i + 8*laneHi), column n = lane%16.
    #pragma unroll
    for (int i = 0; i < 8; ++i) {
        const int m = rowTile * 16 + i + 8 * laneHi;
        const float v = acc[i];
        if (n < 256) {
            Qws[(size_t)m * 256 + n] = v;
        } else if (n < 320) {
            Kws[(size_t)m * 64 + (n - 256)] = v;
        } else if (n < N_PROJ) {
            Gws[(size_t)m * N_HEADS + (n - 320)] = sigmoidf_fast(v + bw[n - 320]);
        }
    }
}

// ---------------------------------------------------------------------------
// Kernel 2: per 16x16 output tile: 4 heads x (K=64) QK^T via f32 WMMA,
// fused sigmoid gating, weighted head-sum, causal mask, NT stores.
// Fully-masked tiles (kt > qt) skip all compute.
// ---------------------------------------------------------------------------
__global__ void __launch_bounds__(256)
gli_score_kernel(const float* __restrict__ Qws,
                 const float* __restrict__ Kws,
                 const float* __restrict__ Gws,
                 const float* __restrict__ head_bias,
                 float* __restrict__ out)
{
    const int wave   = blockIdx.x * (blockDim.x >> 5) + (threadIdx.x >> 5);
    const int lane   = threadIdx.x & 31;
    const int laneM  = lane & 15;
    const int laneHi = lane >> 4;

    const int b   = wave >> 16;         // 256*256 tiles per batch
    const int rem = wave & 0xFFFF;
    const int qt  = rem >> 8;
    const int kt  = rem & 255;

    float* const outBase = out + (size_t)b * SEQ * SEQ;
    const int kcol = kt * 16 + laneM;   // output column owned by this lane

    if (kt > qt) {                      // wave-uniform: tile fully above diagonal
        #pragma unroll
        for (int i = 0; i < 8; ++i) {
            const int q = qt * 16 + i + 8 * laneHi;
            __builtin_nontemporal_store(-__builtin_inff(),
                                        outBase + (size_t)q * SEQ + kcol);
        }
        return;
    }

    const float* qrow = Qws + (size_t)(b * SEQ + qt * 16 + laneM) * 256;
    const float* krow = Kws + (size_t)(b * SEQ + kt * 16 + laneM) * 64;

    float fin[8] = {0.f, 0.f, 0.f, 0.f, 0.f, 0.f, 0.f, 0.f};

    #pragma unroll
    for (int h = 0; h < N_HEADS; ++h) {
        v8f acc = {};
        #pragma unroll
        for (int k0 = 0; k0 < D_IDX; k0 += 4) {
            const int kb = k0 + 2 * laneHi;
            v2f a  = *(const v2f*)(qrow + h * D_IDX + kb);
            v2f bm = *(const v2f*)(krow + kb);
            acc = __builtin_amdgcn_wmma_f32_16x16x4_f32(
                false, a, false, bm, (short)0, acc, false, false);
        }
        const float hb = head_bias[h];
        #pragma unroll
        for (int i = 0; i < 8; ++i) {
            const int qg = b * SEQ + qt * 16 + i + 8 * laneHi;
            const float w = Gws[(size_t)qg * N_HEADS + h];
            fin[i] += sigmoidf_fast(acc[i] * QK_SCALE + hb) * w;
        }
    }

    #pragma unroll
    for (int i = 0; i < 8; ++i) {
        const int q = qt * 16 + i + 8 * laneHi;
        const float v = (kcol > q) ? -__builtin_inff() : fin[i];
        __builtin_nontemporal_store(v, outBase + (size_t)q * SEQ + kcol);
    }
}

// ---------------------------------------------------------------------------
extern "C" void kernel_launch(void* const* d_in, const int* in_sizes, int n_in,
                              void* d_out, int out_size, void* d_ws, size_t ws_size,
                              hipStream_t stream) {
    const float* X   = (const float*)d_in[0];   // [2,4096,2048]
    const float* Wq  = (const float*)d_in[1];   // [256,2048]
    const float* Wk  = (const float*)d_in[2];   // [64,2048]
    const float* Ww  = (const float*)d_in[3];   // [4,2048]
    const float* bw  = (const float*)d_in[4];   // [4]
    const float* hb  = (const float*)d_in[5];   // [4]
    float* out = (float*)d_out;                 // [2,4096,4096]

    float* Qws = (float*)d_ws;                  // 8192*256 f32 = 8 MB
    float* Kws = Qws + (size_t)ROWS * 256;      // 8192*64  f32 = 2 MB
    float* Gws = Kws + (size_t)ROWS * 64;       // 8192*4   f32 = 128 KB

    // Kernel 1: 512 row-tiles x 21 col-tiles = 10752 waves, 8 waves/block.
    {
        const int waves  = (ROWS / 16) * COL_TILES;
        const int blocks = (waves + 7) / 8;
        gli_proj_kernel<<<blocks, 256, 0, stream>>>(X, Wq, Wk, Ww, bw, Qws, Kws, Gws);
    }

    // Kernel 2: 2 * 256 * 256 = 131072 tiles (one wave each), 8 waves/block.
    {
        const int blocks = (BATCH * 256 * 256) / 8;
        gli_score_kernel<<<blocks, 256, 0, stream>>>(Qws, Kws, Gws, hb, out);
    }
}